// DistMult_73985106641032
// MI455X (gfx1250) — compile-verified
//
#include <hip/hip_runtime.h>

// DistMult scoring: out[b] = sum_d src[b,d] * M[rel_idx,d] * dst[b,d]
// B = 500000 rows, D = 128 (per reference), f32 throughout.
//
// Memory-bound: ~514 MB moved -> ~22 us floor at 23.3 TB/s. Strategy:
//  - coalesced float4 (b128) streaming loads, one wave per 16-row tile,
//    branch-free row clamp so phase 1 is one straight-line load/FMA block
//  - per-lane f32 partials -> padded LDS transpose (conflict-free)
//  - cross-lane reduction via V_WMMA_F32_16X16X4_F32 with A = ones:
//      D[m][n] = sum_k B[k][n]  (accumulated over 8 chunks = 32 partials)
//    Every row of D is identical, so only the documented N<->lane mapping
//    matters; robust to K/M layout details.

typedef __attribute__((ext_vector_type(2))) float v2f;
typedef __attribute__((ext_vector_type(8))) float v8f;

#define WAVES_PER_BLOCK 8
#define ROWS_PER_WAVE   16
#define LDS_STRIDE      33   // 32 partials + 1 pad dword -> bank-conflict-free
#define DIM             128

__global__ __launch_bounds__(256)
void distmult_score_kernel(const float* __restrict__ src,
                           const float* __restrict__ dst,
                           const float* __restrict__ M,
                           const int*   __restrict__ rel_idx_p,
                           float*       __restrict__ out,
                           int nrows, int ntiles)
{
    __shared__ float lds[WAVES_PER_BLOCK][ROWS_PER_WAVE * LDS_STRIDE];

    const int lane  = threadIdx.x & 31;
    const int wave  = threadIdx.x >> 5;
    const int gwave = blockIdx.x * WAVES_PER_BLOCK + wave;
    const int nwave = gridDim.x * WAVES_PER_BLOCK;

    // Relation diagonal: lane covers dims [4*lane, 4*lane+3], loaded once.
    const int    ridx = rel_idx_p[0];
    const float4 r4   = *reinterpret_cast<const float4*>(
                            M + (size_t)ridx * DIM + lane * 4);

    float* wlds = lds[wave];

    for (int tile = gwave; tile < ntiles; tile += nwave) {
        const int rowBase = tile * ROWS_PER_WAVE;

        // Phase 1: coalesced 512B/row loads; one f32 partial per lane/row.
        // Branch-free: clamp row so tail-tile loads stay in bounds; the
        // clamped rows' partials are computed but never stored.
        #pragma unroll
        for (int m = 0; m < ROWS_PER_WAVE; ++m) {
            int row = rowBase + m;                  // wave-uniform
            row = row < nrows ? row : nrows - 1;    // s_min, no EXEC churn
            const size_t off = (size_t)row * DIM + lane * 4;
            const float4 s4 = *reinterpret_cast<const float4*>(src + off);
            const float4 d4 = *reinterpret_cast<const float4*>(dst + off);
            const float p = s4.x * r4.x * d4.x + s4.y * r4.y * d4.y +
                            s4.z * r4.z * d4.z + s4.w * r4.w * d4.w;
            wlds[m * LDS_STRIDE + lane] = p;        // consecutive -> no conflict
        }
        __builtin_amdgcn_wave_barrier();            // order LDS W->R in-wave

        // Phase 2: WMMA reduction. B[k][n] = p[row n][4*ch + (k>>1) + 2*(k&1)]; A = ones.
        const int r = lane & 15;
        const int g = lane >> 4;
        v2f a; a.x = 1.0f; a.y = 1.0f;              // 16x4 all-ones A matrix
        v8f c = {};
        #pragma unroll
        for (int ch = 0; ch < 8; ++ch) {
            v2f b;
            b.x = wlds[r * LDS_STRIDE + 4 * ch + g];       // K = 2*g slot
            b.y = wlds[r * LDS_STRIDE + 4 * ch + 2 + g];   // K = 2*g+1 slot
            c = __builtin_amdgcn_wmma_f32_16x16x4_f32(
                    /*neg_a=*/false, a, /*neg_b=*/false, b,
                    /*c_mod=*/(short)0, c,
                    /*reuse_a=*/false, /*reuse_b=*/false);
        }
        __builtin_amdgcn_wave_barrier();            // reads done before next W

        // Every D row identical: VGPR0 of lane L holds rowsum[L & 15].
        const int orow = rowBase + r;
        if (g == 0 && orow < nrows) out[orow] = c[0];
    }
}

extern "C" void kernel_launch(void* const* d_in, const int* in_sizes, int n_in,
                              void* d_out, int out_size, void* d_ws, size_t ws_size,
                              hipStream_t stream)
{
    const float* src  = (const float*)d_in[0];
    const float* dst  = (const float*)d_in[1];
    const float* M    = (const float*)d_in[2];
    const int*   ridx = (const int*)d_in[3];
    float*       out  = (float*)d_out;

    const int nrows  = in_sizes[0] / DIM;
    const int ntiles = (nrows + ROWS_PER_WAVE - 1) / ROWS_PER_WAVE;

    int blocks = (ntiles + WAVES_PER_BLOCK - 1) / WAVES_PER_BLOCK;
    if (blocks > 4096) blocks = 4096;
    if (blocks < 1)    blocks = 1;

    distmult_score_kernel<<<blocks, 256, 0, stream>>>(
        src, dst, M, ridx, out, nrows, ntiles);
}